// AttentiveAtlasEncoder_36318243455407
// MI455X (gfx1250) — compile-verified
//
#include <hip/hip_runtime.h>
#include <hip/hip_bf16.h>
#include <math.h>

typedef __attribute__((ext_vector_type(2))) float v2f;
typedef __attribute__((ext_vector_type(4))) float v4f;
typedef __attribute__((ext_vector_type(8))) float v8f;
typedef __attribute__((ext_vector_type(4))) unsigned int v4u;
typedef __attribute__((ext_vector_type(8))) int v8i;
typedef __attribute__((ext_vector_type(4))) int v4i;

#define NB      131072
#define IN_DIMS 32
#define HIDS    128
#define DD      16
#define NCH     8
#define KCODE   128
#define SFH     8

#define WPB     8            // waves per block (phase-aligned via barriers)
#define ROWS_PB (WPB * 16)   // 128 rows per block

#define TSTRIDE 132   // per-wave 16x128 tile stride: 132%64=4 -> conflict-free, even -> b64 aligned
#define VSTRIDE 18    // v tile stride: even (b64 align), 18r%64 distinct for r<16
#define W2S     136   // staged-W2 LDS stride: 2*136%64=16 -> half-waves hit disjoint bank windows

// flat float output offsets, in reference tuple order
#define OFF_KCHART ((size_t)0)
#define OFF_KCODE  ((size_t)NB)
#define OFF_ZN     ((size_t)2*NB)
#define OFF_ZTEX   ((size_t)18*NB)
#define OFF_RW     ((size_t)34*NB)
#define OFF_ZGEO   ((size_t)42*NB)
#define OFF_VQ     ((size_t)58*NB)
#define OFF_IDX    ((size_t)58*NB + 1)
#define OFF_ZNALL  ((size_t)66*NB + 1)
#define OFF_CBAR   ((size_t)194*NB + 1)

__device__ __forceinline__ v8f wmma4(v2f a, v2f b, v8f c) {
  // V_WMMA_F32_16X16X4_F32 : fp32 tensor path, matches fp32 reference numerics
  return __builtin_amdgcn_wmma_f32_16x16x4_f32(false, a, false, b, (short)0, c,
                                               false, false);
}

__device__ __forceinline__ float gelu_exact(float x) {
  return 0.5f * x * (1.0f + erff(x * 0.7071067811865475f));
}

// Tensor Data Mover: 2D tile load Global->LDS, data_size=4B, optional LDS pad.
// pad codes per D#: interval 2^(c+1) DWORDs, amount (c+1) DWORDs.
__device__ __forceinline__ void tdm_load_2d(const void* gaddr, unsigned lds_off,
                                            unsigned dim0, unsigned dim1,
                                            unsigned tile0, unsigned tile1,
                                            unsigned stride0,
                                            int pad_en, unsigned pad_i, unsigned pad_a) {
  unsigned long long ga = (unsigned long long)gaddr;
  v4u g0;
  g0[0] = 1u;                                              // count=1, user D#
  g0[1] = lds_off;                                         // LDS byte address
  g0[2] = (unsigned)(ga & 0xFFFFFFFFu);                    // global_addr[31:0]
  g0[3] = (unsigned)((ga >> 32) & 0x1FFFFFFu) | (2u << 30);// global_addr[56:32] | type=2
  v8i g1;
  unsigned w0 = (2u << 16);                                // data_size = 4B, no multicast
  if (pad_en) w0 |= (1u << 20) | (pad_i << 22) | (pad_a << 25);
  g1[0] = (int)w0;
  g1[1] = (int)(dim0 << 16);                               // tensor_dim0[15:0]
  g1[2] = (int)((dim0 >> 16) | (dim1 << 16));              // tensor_dim0[31:16] | tensor_dim1[15:0]
  g1[3] = (int)((dim1 >> 16) | (tile0 << 16));             // tensor_dim1[31:16] | tile_dim0
  g1[4] = (int)(tile1 & 0xFFFFu);                          // tile_dim1 (tile_dim2=0)
  g1[5] = (int)stride0;                                    // tensor_dim0_stride[31:0]
  g1[6] = 0;
  g1[7] = 0;
  v4i gz = {0, 0, 0, 0};
#if __clang_major__ >= 23
  v8i gz8 = {0, 0, 0, 0, 0, 0, 0, 0};
  __builtin_amdgcn_tensor_load_to_lds(g0, g1, gz, gz, gz8, 0);
#else
  __builtin_amdgcn_tensor_load_to_lds(g0, g1, gz, gz, 0);
#endif
}

__global__ __launch_bounds__(WPB * 32) void atlas_encoder_kernel(
    const float* __restrict__ x,  const float* __restrict__ W1, const float* __restrict__ b1,
    const float* __restrict__ W2, const float* __restrict__ b2,
    const float* __restrict__ Wv, const float* __restrict__ bv,
    const float* __restrict__ cc, const float* __restrict__ cb,
    const float* __restrict__ sfW1, const float* __restrict__ sfb1,
    const float* __restrict__ sfW2, const float* __restrict__ sfb2,
    float* __restrict__ out)
{
  __shared__ float s_w2 [HIDS * W2S];          // W2 staged by TDM, padded stride 136
  __shared__ float s_wv [HIDS * DD];           // Wv staged by TDM
  __shared__ float s_cn2[NCH * KCODE];         // codebook squared norms
  __shared__ float s_tile[WPB * 16 * TSTRIDE]; // per-wave h / dot staging
  __shared__ float s_vv [WPB * 16 * VSTRIDE];  // per-wave v then v_local (in-place)
  __shared__ float s_red[WPB];

  const int tid  = (int)threadIdx.x;
  const int lane = tid & 31;
  const int wid  = tid >> 5;
  const int lo   = lane & 15;
  const int hi   = lane >> 4;
  const int row0 = (int)blockIdx.x * ROWS_PB + wid * 16;

  float* tile = s_tile + wid * 16 * TSTRIDE;
  float* vv   = s_vv   + wid * 16 * VSTRIDE;

  const v8f vzero = {0.f,0.f,0.f,0.f,0.f,0.f,0.f,0.f};

  // ---- kick off TDM weight staging (wave 0 only; scalar branch since TDM ignores EXEC) ----
  if (__builtin_amdgcn_readfirstlane(wid) == 0) {
    // W2 [128x128] -> LDS stride 136 (pad 8 dwords every 128 dwords: codes 6 / 7)
    tdm_load_2d(W2, (unsigned)(uintptr_t)&s_w2[0], HIDS, HIDS, HIDS, HIDS, HIDS, 1, 6u, 7u);
    // Wv [128x16] -> LDS stride 16 (no pad needed: 2*16%64=32 -> disjoint windows)
    tdm_load_2d(Wv, (unsigned)(uintptr_t)&s_wv[0], DD, HIDS, DD, HIDS, DD, 0, 0u, 0u);
  }

  // ---- codebook squared norms -> LDS (overlapped with TDM) ----
  for (int i = tid; i < NCH * KCODE; i += WPB * 32) {
    const float* c = cb + (size_t)i * DD;
    float s = 0.f;
    #pragma unroll
    for (int d = 0; d < DD; ++d) s += c[d] * c[d];
    s_cn2[i] = s;
  }

  // ---- GEMM1: h = gelu(x @ W1 + b1), [16x32]@[32x128], fp32 WMMA (W1 from L2) ----
  v8f acc[8];
  #pragma unroll
  for (int t = 0; t < 8; ++t) acc[t] = vzero;
  for (int k = 0; k < IN_DIMS / 4; ++k) {
    const float* xa = x + (size_t)(row0 + lo) * IN_DIMS + 4 * k + 2 * hi;
    v2f a; a[0] = xa[0]; a[1] = xa[1];
    #pragma unroll
    for (int t = 0; t < 8; ++t) {
      const int col = 16 * t + lo;
      v2f bf;
      bf[0] = W1[(size_t)(4 * k + 2 * hi)     * HIDS + col];
      bf[1] = W1[(size_t)(4 * k + 2 * hi + 1) * HIDS + col];
      acc[t] = wmma4(a, bf, acc[t]);
    }
  }
  #pragma unroll
  for (int t = 0; t < 8; ++t) {
    const int col  = 16 * t + lo;
    const float bb = b1[col];
    #pragma unroll
    for (int r = 0; r < 8; ++r)
      tile[(r + 8 * hi) * TSTRIDE + col] = gelu_exact(acc[t][r] + bb);
  }

  // wave 0 drains its TENSORcnt before releasing the block
  if (__builtin_amdgcn_readfirstlane(wid) == 0)
    __builtin_amdgcn_s_wait_tensorcnt(0);
  __syncthreads();

  // ---- GEMM2: h2 = gelu(h @ W2 + b2), [16x128]@[128x128], W2 from LDS ----
  v8f acc2[8];
  #pragma unroll
  for (int t = 0; t < 8; ++t) acc2[t] = vzero;
  for (int k = 0; k < HIDS / 4; ++k) {
    v2f a = *(const v2f*)&tile[lo * TSTRIDE + 4 * k + 2 * hi];   // 8B aligned
    #pragma unroll
    for (int t = 0; t < 8; ++t) {
      const int col = 16 * t + lo;
      v2f bf;
      bf[0] = s_w2[(4 * k + 2 * hi)     * W2S + col];
      bf[1] = s_w2[(4 * k + 2 * hi + 1) * W2S + col];
      acc2[t] = wmma4(a, bf, acc2[t]);
    }
  }
  __syncthreads();
  #pragma unroll
  for (int t = 0; t < 8; ++t) {
    const int col  = 16 * t + lo;
    const float bb = b2[col];
    #pragma unroll
    for (int r = 0; r < 8; ++r)
      tile[(r + 8 * hi) * TSTRIDE + col] = gelu_exact(acc2[t][r] + bb);
  }
  __syncthreads();

  // ---- GEMM3: v = h2 @ Wv + bv, [16x128]@[128x16], Wv from LDS ----
  v8f accv = vzero;
  for (int k = 0; k < HIDS / 4; ++k) {
    v2f a = *(const v2f*)&tile[lo * TSTRIDE + 4 * k + 2 * hi];
    v2f bf;
    bf[0] = s_wv[(4 * k + 2 * hi)     * DD + lo];
    bf[1] = s_wv[(4 * k + 2 * hi + 1) * DD + lo];
    accv = wmma4(a, bf, accv);
  }
  {
    const float bb = bv[lo];
    #pragma unroll
    for (int r = 0; r < 8; ++r)
      vv[(r + 8 * hi) * VSTRIDE + lo] = accv[r] + bb;
  }
  __syncthreads();

  // ---- per-row router / softmax / v_local (lanes 0..15 of each wave) ----
  float rw[NCH], cbar[DD], vloc[DD];
  int   kchart = 0;
  float loss_acc = 0.f;
  if (lane < 16) {
    float v[DD];
    #pragma unroll
    for (int d = 0; d < DD; ++d) v[d] = vv[lane * VSTRIDE + d];
    float sc[NCH], smax = -3.4e38f;
    #pragma unroll
    for (int n = 0; n < NCH; ++n) {
      float s = 0.f;
      #pragma unroll
      for (int d = 0; d < DD; ++d) s += v[d] * cc[n * DD + d];
      s *= 0.25f;                    // 1/sqrt(16)
      sc[n] = s;
      if (s > smax) smax = s;
    }
    float bm = sc[0];
    #pragma unroll
    for (int n = 1; n < NCH; ++n)    // first-occurrence argmax
      if (sc[n] > bm) { bm = sc[n]; kchart = n; }
    float esum = 0.f;
    #pragma unroll
    for (int n = 0; n < NCH; ++n) { float e = __expf(sc[n] - smax); rw[n] = e; esum += e; }
    const float inv = 1.f / esum;
    #pragma unroll
    for (int n = 0; n < NCH; ++n) rw[n] *= inv;
    #pragma unroll
    for (int d = 0; d < DD; ++d) {
      float cv = 0.f;
      #pragma unroll
      for (int n = 0; n < NCH; ++n) cv += rw[n] * cc[n * DD + d];
      cbar[d] = cv;
      vloc[d] = v[d] - cv;
      vv[lane * VSTRIDE + d] = vloc[d];  // in-place: row's v already in registers
    }
  }
  __syncthreads();

  // ---- VQ: per chart, dots via WMMA, split argmin scan, chart MLP ----
  float zqb[DD], zn[DD];
  #pragma unroll
  for (int d = 0; d < DD; ++d) { zqb[d] = 0.f; zn[d] = 0.f; }
  int kcode_sel = 0;

  for (int n = 0; n < NCH; ++n) {
    // dot[r][c] = v_local[r] . codebook[n][c] : [16x16]@[16x128] fp32 WMMA
    #pragma unroll
    for (int t = 0; t < 8; ++t) {
      v8f dacc = vzero;
      #pragma unroll
      for (int k = 0; k < 4; ++k) {
        v2f a  = *(const v2f*)&vv[lo * VSTRIDE + 4 * k + 2 * hi];
        v2f bf = *(const v2f*)&cb[(((size_t)n * KCODE + 16 * t + lo) * DD) + 4 * k + 2 * hi];
        dacc = wmma4(a, bf, dacc);
      }
      #pragma unroll
      for (int r = 0; r < 8; ++r)
        tile[(r + 8 * hi) * TSTRIDE + 16 * t + lo] = dacc[r];
    }
    __syncthreads();

    // argmin over ||c||^2 - 2 v.c, split across half-waves:
    // hi=0 scans codes 0..63, hi=1 scans 64..127 rotated by 32 (bank-disjoint:
    // column delta 96 == 32 mod 64). Explicit (value,index) tie-break keeps
    // jnp.argmin first-occurrence semantics under the reordering.
    float best = 3.4e38f; int bi = hi * 64;
    for (int k2 = 0; k2 < 64; ++k2) {
      const int idx = hi ? (64 + ((k2 + 32) & 63)) : k2;
      const float d2 = s_cn2[n * KCODE + idx] - 2.f * tile[lo * TSTRIDE + idx];
      if (d2 < best || (d2 == best && idx < bi)) { best = d2; bi = idx; }
    }
    {
      const float ob = __shfl_xor(best, 16);
      const int   oi = __shfl_xor(bi, 16);
      if (ob < best || (ob == best && oi < bi)) { best = ob; bi = oi; }
    }

    if (lane < 16) {
      const size_t gr = (size_t)(row0 + lane);
      const float* zq  = cb + ((size_t)n * KCODE + bi) * DD;
      const float  rwn = rw[n];
      float delta[DD];
      #pragma unroll
      for (int d = 0; d < DD; ++d) {
        const float dl = vloc[d] - zq[d];
        delta[d] = dl;
        loss_acc += dl * dl * rwn;           // codebook + 0.25*commitment share fwd value
      }
      float hsf[SFH];
      #pragma unroll
      for (int j = 0; j < SFH; ++j) {
        float s = sfb1[j];
        #pragma unroll
        for (int d = 0; d < DD; ++d) s += delta[d] * sfW1[d * SFH + j];
        hsf[j] = gelu_exact(s);
      }
      #pragma unroll
      for (int d = 0; d < DD; ++d) {
        float s = sfb2[d];
        #pragma unroll
        for (int j = 0; j < SFH; ++j) s += hsf[j] * sfW2[j * DD + d];
        out[OFF_ZNALL + ((size_t)gr * NCH + n) * DD + d] = s;  // +1 offset: 4B aligned only
        zn[d]  += rwn * s;
        zqb[d] += rwn * zq[d];
      }
      out[OFF_IDX + gr * NCH + n] = (float)bi;
      if (n == kchart) kcode_sel = bi;
    }
    __syncthreads();
  }

  // ---- final per-row outputs (b128 stores; ZN/ZTEX/ZGEO/RW 16B-aligned, CBAR only 4B) ----
  if (lane < 16) {
    const size_t gr = (size_t)(row0 + lane);
    out[OFF_KCHART + gr] = (float)kchart;
    out[OFF_KCODE  + gr] = (float)kcode_sel;
    #pragma unroll
    for (int dq = 0; dq < 4; ++dq) {
      v4f qn, qt, qg, qc;
      #pragma unroll
      for (int e = 0; e < 4; ++e) {
        const int d = 4 * dq + e;
        const float dblend = vloc[d] - zqb[d];
        qn[e] = zn[d];
        qt[e] = dblend - zn[d];
        qg[e] = cbar[d] + zqb[d] + zn[d];    // z_q_st fwd == z_q_blended
        qc[e] = cbar[d];
      }
      *(v4f*)&out[OFF_ZN   + gr * DD + 4 * dq] = qn;
      *(v4f*)&out[OFF_ZTEX + gr * DD + 4 * dq] = qt;
      *(v4f*)&out[OFF_ZGEO + gr * DD + 4 * dq] = qg;
      *(v4f*)&out[OFF_CBAR + gr * DD + 4 * dq] = qc;
    }
    #pragma unroll
    for (int nq = 0; nq < 2; ++nq) {
      v4f qr;
      #pragma unroll
      for (int e = 0; e < 4; ++e) qr[e] = rw[4 * nq + e];
      *(v4f*)&out[OFF_RW + gr * NCH + 4 * nq] = qr;
    }
  }

  // ---- vq_loss: wave shuffle-reduce, block LDS-reduce, one atomic per block ----
  float l = loss_acc;
  l += __shfl_xor(l, 1);
  l += __shfl_xor(l, 2);
  l += __shfl_xor(l, 4);
  l += __shfl_xor(l, 8);
  l += __shfl_xor(l, 16);
  if (lane == 0) s_red[wid] = l;
  __syncthreads();
  if (tid == 0) {
    float t = 0.f;
    #pragma unroll
    for (int w = 0; w < WPB; ++w) t += s_red[w];
    atomicAdd(&out[OFF_VQ], t * (1.25f / ((float)NB * (float)DD)));
  }
}

extern "C" void kernel_launch(void* const* d_in, const int* in_sizes, int n_in,
                              void* d_out, int out_size, void* d_ws, size_t ws_size,
                              hipStream_t stream) {
  const float* x    = (const float*)d_in[0];
  const float* W1   = (const float*)d_in[1];
  const float* b1   = (const float*)d_in[2];
  const float* W2   = (const float*)d_in[3];
  const float* b2   = (const float*)d_in[4];
  const float* Wv   = (const float*)d_in[5];
  const float* bv   = (const float*)d_in[6];
  const float* cc   = (const float*)d_in[7];
  const float* cb   = (const float*)d_in[8];
  const float* sfW1 = (const float*)d_in[9];
  const float* sfb1 = (const float*)d_in[10];
  const float* sfW2 = (const float*)d_in[11];
  const float* sfb2 = (const float*)d_in[12];
  float* out = (float*)d_out;

  // zero the vq_loss accumulator slot (graph-capture safe)
  hipMemsetAsync((void*)(out + OFF_VQ), 0, sizeof(float), stream);

  dim3 grid(NB / ROWS_PB), block(WPB * 32);
  hipLaunchKernelGGL(atlas_encoder_kernel, grid, block, 0, stream,
                     x, W1, b1, W2, b2, Wv, bv, cc, cb, sfW1, sfb1, sfW2, sfb2, out);
}